// QuantumAttention_36842229465265
// MI455X (gfx1250) — compile-verified
//
#include <hip/hip_runtime.h>
#include <hip/hip_bf16.h>
#include <math.h>

typedef __attribute__((ext_vector_type(2))) float v2f;
typedef __attribute__((ext_vector_type(8))) float v8f;

#define PI_F 3.14159265358979323846f

// ---------------------------------------------------------------------------
// Pre-net GEMM: Y[B,4] = X[B,96] @ W[4,96]^T + b, via V_WMMA_F32_16X16X4_F32.
// One wave per 16-row tile.  A: 16x4 f32 (2 VGPRs), B: 4x16 f32 (2 VGPRs),
// C/D: 16x16 f32 (8 VGPRs).  K=96 in 24 steps of 4.
// ---------------------------------------------------------------------------
__global__ __launch_bounds__(256) void qattn_prenet_wmma(
    const float* __restrict__ X, const float* __restrict__ W,
    const float* __restrict__ bias, float* __restrict__ Y, int B) {
  int wave = (blockIdx.x * blockDim.x + threadIdx.x) >> 5;
  int lane = threadIdx.x & 31;
  int row0 = wave * 16;
  if (row0 >= B) return;                 // wave-uniform exit

  int half = lane >> 4;                  // 0: K pair (k,k+1); 1: (k+2,k+3)
  int l    = lane & 15;

  // B-matrix: Bmat[k][n] = W[n*96+k], valid for n<4; mask higher lanes to 0
  // without divergence (EXEC must be all-ones at each WMMA).
  int   n   = (l < 4) ? l : 3;
  float msk = (l < 4) ? 1.f : 0.f;
  const float* __restrict__ xrow = X + (size_t)(row0 + l) * 96;
  const float* __restrict__ wrow = W + (size_t)n * 96;

  v8f acc = {};
#pragma unroll
  for (int k0 = 0; k0 < 96; k0 += 4) {
    v2f a, b;
    a.x = xrow[k0 + 2 * half + 0];
    a.y = xrow[k0 + 2 * half + 1];
    b.x = wrow[k0 + 2 * half + 0] * msk;
    b.y = wrow[k0 + 2 * half + 1] * msk;
    acc = __builtin_amdgcn_wmma_f32_16x16x4_f32(
        false, a, false, b, (short)0, acc, false, false);
  }
  // C/D layout: lanes 0-15 -> (M=v, N=l); lanes 16-31 -> (M=v+8, N=l)
  if (l < 4) {
    float bv = bias[l];
#pragma unroll
    for (int v = 0; v < 8; ++v) {
      int m = v + half * 8;
      Y[(size_t)(row0 + m) * 4 + l] = acc[v] + bv;
    }
  }
}

// ---------------------------------------------------------------------------
// Wave-level helpers
// ---------------------------------------------------------------------------
__device__ __forceinline__ float wave_sum(float v) {
#pragma unroll
  for (int m = 16; m >= 1; m >>= 1) v += __shfl_xor(v, m, 32);
  return v;
}

// ======================= 8-qubit state (256 amps) ==========================
// amp index bits [4:0] = lane, bits [7:5] = register r (qubit w = bit w).
// ===========================================================================
template <int W>
__device__ __forceinline__ void rz8(float (&sr)[8], float (&si)[8], int lane, float t) {
  float s, c; __sincosf(0.5f * t, &s, &c);
  if constexpr (W < 5) {
    float sg = ((lane >> W) & 1) ? s : -s;
#pragma unroll
    for (int r = 0; r < 8; ++r) {
      float re = sr[r], im = si[r];
      sr[r] = c * re - sg * im;
      si[r] = c * im + sg * re;
    }
  } else {
#pragma unroll
    for (int r = 0; r < 8; ++r) {
      float sg = ((r >> (W - 5)) & 1) ? s : -s;
      float re = sr[r], im = si[r];
      sr[r] = c * re - sg * im;
      si[r] = c * im + sg * re;
    }
  }
}

template <int W>
__device__ __forceinline__ void rx8(float (&sr)[8], float (&si)[8], float t) {
  float s, c; __sincosf(0.5f * t, &s, &c);
  if constexpr (W < 5) {
#pragma unroll
    for (int r = 0; r < 8; ++r) {
      float pr = __shfl_xor(sr[r], 1 << W, 32);
      float pi = __shfl_xor(si[r], 1 << W, 32);
      float ar = sr[r], ai = si[r];
      sr[r] = c * ar + s * pi;   // new = c*self + (-i s)*partner
      si[r] = c * ai - s * pr;
    }
  } else {
    constexpr int b = 1 << (W - 5);
#pragma unroll
    for (int r = 0; r < 8; ++r)
      if (!(r & b)) {
        int p = r | b;
        float a0r = sr[r], a0i = si[r], a1r = sr[p], a1i = si[p];
        sr[r] = c * a0r + s * a1i;  si[r] = c * a0i - s * a1r;
        sr[p] = c * a1r + s * a0i;  si[p] = c * a1i - s * a0r;
      }
  }
}

template <int CW, int TW>
__device__ __forceinline__ void crx8(float (&sr)[8], float (&si)[8], int lane, float t) {
  float s, c; __sincosf(0.5f * t, &s, &c);
  if constexpr (TW < 5) {
#pragma unroll
    for (int r = 0; r < 8; ++r) {
      float pr = __shfl_xor(sr[r], 1 << TW, 32);
      float pi = __shfl_xor(si[r], 1 << TW, 32);
      bool act;
      if constexpr (CW < 5) act = (lane >> CW) & 1; else act = (r >> (CW - 5)) & 1;
      float ar = sr[r], ai = si[r];
      float nr = c * ar + s * pi, ni = c * ai - s * pr;
      sr[r] = act ? nr : ar;
      si[r] = act ? ni : ai;
    }
  } else {
    constexpr int b = 1 << (TW - 5);
#pragma unroll
    for (int r = 0; r < 8; ++r)
      if (!(r & b)) {
        int p = r | b;
        bool act;
        if constexpr (CW < 5) act = (lane >> CW) & 1; else act = (r >> (CW - 5)) & 1;
        float a0r = sr[r], a0i = si[r], a1r = sr[p], a1i = si[p];
        float n0r = c * a0r + s * a1i, n0i = c * a0i - s * a1r;
        float n1r = c * a1r + s * a0i, n1i = c * a1i - s * a0r;
        sr[r] = act ? n0r : a0r; si[r] = act ? n0i : a0i;
        sr[p] = act ? n1r : a1r; si[p] = act ? n1i : a1i;
      }
  }
}

template <int CW, int TW>
__device__ __forceinline__ void cnot8(float (&sr)[8], float (&si)[8], int lane) {
  if constexpr (TW < 5) {
#pragma unroll
    for (int r = 0; r < 8; ++r) {
      float pr = __shfl_xor(sr[r], 1 << TW, 32);
      float pi = __shfl_xor(si[r], 1 << TW, 32);
      bool act;
      if constexpr (CW < 5) act = (lane >> CW) & 1; else act = (r >> (CW - 5)) & 1;
      sr[r] = act ? pr : sr[r];
      si[r] = act ? pi : si[r];
    }
  } else {
    constexpr int b = 1 << (TW - 5);
#pragma unroll
    for (int r = 0; r < 8; ++r)
      if (!(r & b)) {
        int p = r | b;
        bool act;
        if constexpr (CW < 5) act = (lane >> CW) & 1; else act = (r >> (CW - 5)) & 1;
        float a0r = sr[r], a0i = si[r], a1r = sr[p], a1i = si[p];
        sr[r] = act ? a1r : a0r; si[r] = act ? a1i : a0i;
        sr[p] = act ? a0r : a1r; si[p] = act ? a0i : a1i;
      }
  }
}

template <int OFF>
__device__ __forceinline__ void init_qkv8(float (&sr)[8], float (&si)[8], int lane,
                                          const float* __restrict__ rot,
                                          const float* __restrict__ crx) {
  rz8<OFF + 0>(sr, si, lane, rot[0]);  rx8<OFF + 0>(sr, si, rot[1]);  rz8<OFF + 0>(sr, si, lane, rot[2]);
  rz8<OFF + 1>(sr, si, lane, rot[3]);  rx8<OFF + 1>(sr, si, rot[4]);  rz8<OFF + 1>(sr, si, lane, rot[5]);
  rz8<OFF + 2>(sr, si, lane, rot[6]);  rx8<OFF + 2>(sr, si, rot[7]);  rz8<OFF + 2>(sr, si, lane, rot[8]);
  rz8<OFF + 3>(sr, si, lane, rot[9]);  rx8<OFF + 3>(sr, si, rot[10]); rz8<OFF + 3>(sr, si, lane, rot[11]);
  crx8<OFF + 0, OFF + 1>(sr, si, lane, crx[0]);
  crx8<OFF + 1, OFF + 2>(sr, si, lane, crx[1]);
  crx8<OFF + 2, OFF + 3>(sr, si, lane, crx[2]);
  crx8<OFF + 3, OFF + 0>(sr, si, lane, crx[3]);
  cnot8<OFF + 0, OFF + 1>(sr, si, lane);
  cnot8<OFF + 1, OFF + 2>(sr, si, lane);
  cnot8<OFF + 2, OFF + 3>(sr, si, lane);
  cnot8<OFF + 3, OFF + 0>(sr, si, lane);
  cnot8<OFF + 0, OFF + 3>(sr, si, lane);
  cnot8<OFF + 1, OFF + 0>(sr, si, lane);
  cnot8<OFF + 2, OFF + 1>(sr, si, lane);
  cnot8<OFF + 3, OFF + 2>(sr, si, lane);
  rz8<OFF + 0>(sr, si, lane, rot[12]); rx8<OFF + 0>(sr, si, rot[13]); rz8<OFF + 0>(sr, si, lane, rot[14]);
  rz8<OFF + 1>(sr, si, lane, rot[15]); rx8<OFF + 1>(sr, si, rot[16]); rz8<OFF + 1>(sr, si, lane, rot[17]);
  rz8<OFF + 2>(sr, si, lane, rot[18]); rx8<OFF + 2>(sr, si, rot[19]); rz8<OFF + 2>(sr, si, lane, rot[20]);
  rz8<OFF + 3>(sr, si, lane, rot[21]); rx8<OFF + 3>(sr, si, rot[22]); rz8<OFF + 3>(sr, si, lane, rot[23]);
}

// <Z_W> and <X_W> for a lane-bit wire (W < 5)
template <int W>
__device__ __forceinline__ void expzx8(const float (&sr)[8], const float (&si)[8],
                                       int lane, float& z, float& x) {
  float zz = 0.f, xx = 0.f;
  float sgn = ((lane >> W) & 1) ? -1.f : 1.f;
#pragma unroll
  for (int r = 0; r < 8; ++r) {
    zz += sgn * (sr[r] * sr[r] + si[r] * si[r]);
    float pr = __shfl_xor(sr[r], 1 << W, 32);
    float pi = __shfl_xor(si[r], 1 << W, 32);
    xx += sr[r] * pr + si[r] * pi;
  }
  z = wave_sum(zz);
  x = wave_sum(xx);
}

// ======================= 4-qubit state (16 amps) ===========================
// 1 amplitude per lane (lane bits 0..3); lanes 16-31 hold zero.
// ===========================================================================
template <int W>
__device__ __forceinline__ void rz4(float& vr, float& vi, int lane, float t) {
  float s, c; __sincosf(0.5f * t, &s, &c);
  float sg = ((lane >> W) & 1) ? s : -s;
  float re = vr, im = vi;
  vr = c * re - sg * im;
  vi = c * im + sg * re;
}

template <int W>
__device__ __forceinline__ void rx4(float& vr, float& vi, float t) {
  float s, c; __sincosf(0.5f * t, &s, &c);
  float pr = __shfl_xor(vr, 1 << W, 32);
  float pi = __shfl_xor(vi, 1 << W, 32);
  float ar = vr, ai = vi;
  vr = c * ar + s * pi;
  vi = c * ai - s * pr;
}

template <int CW, int TW>
__device__ __forceinline__ void crx4(float& vr, float& vi, int lane, float t) {
  float s, c; __sincosf(0.5f * t, &s, &c);
  float pr = __shfl_xor(vr, 1 << TW, 32);
  float pi = __shfl_xor(vi, 1 << TW, 32);
  bool act = (lane >> CW) & 1;
  float nr = c * vr + s * pi, ni = c * vi - s * pr;
  vr = act ? nr : vr;
  vi = act ? ni : vi;
}

template <int CW, int TW>
__device__ __forceinline__ void cnot4(float& vr, float& vi, int lane) {
  float pr = __shfl_xor(vr, 1 << TW, 32);
  float pi = __shfl_xor(vi, 1 << TW, 32);
  bool act = (lane >> CW) & 1;
  vr = act ? pr : vr;
  vi = act ? pi : vi;
}

template <int W>
__device__ __forceinline__ void expzx4(float vr, float vi, int lane, float& z, float& x) {
  float sgn = ((lane >> W) & 1) ? -1.f : 1.f;
  float zz = sgn * (vr * vr + vi * vi);
  float pr = __shfl_xor(vr, 1 << W, 32);
  float pi = __shfl_xor(vi, 1 << W, 32);
  float xx = vr * pr + vi * pi;
  z = wave_sum(zz);
  x = wave_sum(xx);
}

// ---------------------------------------------------------------------------
// Main simulation kernel: one batch element per wave32.
// ---------------------------------------------------------------------------
__global__ __launch_bounds__(256) void qattn_sim(
    const float* __restrict__ x1p, const float* __restrict__ x2p,
    const float* __restrict__ wq_rot, const float* __restrict__ wq_crx,
    const float* __restrict__ wk_rot, const float* __restrict__ wk_crx,
    const float* __restrict__ wv_rot, const float* __restrict__ wv_crx,
    float* __restrict__ out, int B) {
  int lane = threadIdx.x & 31;
  int wv   = (int)((blockIdx.x * blockDim.x + threadIdx.x) >> 5);
  if (wv >= B) return;                   // wave-uniform exit

  float xq[4], xk[4];
#pragma unroll
  for (int i = 0; i < 4; ++i) {
    xq[i] = x1p[(size_t)wv * 4 + i];
    xk[i] = x2p[(size_t)wv * 4 + i];
  }

  // ---- score_circuit: 8 qubits, |0>^8 ----
  float sr[8], si[8];
#pragma unroll
  for (int r = 0; r < 8; ++r) { sr[r] = 0.f; si[r] = 0.f; }
  sr[0] = (lane == 0) ? 1.f : 0.f;

  rx8<0>(sr, si, xq[0]); rx8<1>(sr, si, xq[1]); rx8<2>(sr, si, xq[2]); rx8<3>(sr, si, xq[3]);
  init_qkv8<0>(sr, si, lane, wq_rot, wq_crx);
  rx8<0>(sr, si, xk[0]); rx8<1>(sr, si, xk[1]); rx8<2>(sr, si, xk[2]); rx8<3>(sr, si, xk[3]);
  init_qkv8<4>(sr, si, lane, wk_rot, wk_crx);
  cnot8<0, 4>(sr, si, lane); cnot8<4, 0>(sr, si, lane);
  cnot8<1, 5>(sr, si, lane); cnot8<5, 1>(sr, si, lane);
  cnot8<2, 6>(sr, si, lane); cnot8<6, 2>(sr, si, lane);
  cnot8<3, 7>(sr, si, lane); cnot8<7, 3>(sr, si, lane);

  float z[4], x[4];
  expzx8<1>(sr, si, lane, z[0], x[0]);
  expzx8<2>(sr, si, lane, z[1], x[1]);
  expzx8<3>(sr, si, lane, z[2], x[2]);
  expzx8<4>(sr, si, lane, z[3], x[3]);

  float mag[4];
#pragma unroll
  for (int i = 0; i < 4; ++i) mag[i] = sqrtf(z[i] * z[i] + x[i] * x[i]);

  // ---- value_circuit: 4 qubits, 1 amp/lane (lanes>=16 zero) ----
  float vr = (lane == 0) ? 1.f : 0.f, vi = 0.f;
  rx4<0>(vr, vi, xk[0]); rx4<1>(vr, vi, xk[1]); rx4<2>(vr, vi, xk[2]); rx4<3>(vr, vi, xk[3]);
  // init_qkv on 4 qubits
  rz4<0>(vr, vi, lane, wv_rot[0]);  rx4<0>(vr, vi, wv_rot[1]);  rz4<0>(vr, vi, lane, wv_rot[2]);
  rz4<1>(vr, vi, lane, wv_rot[3]);  rx4<1>(vr, vi, wv_rot[4]);  rz4<1>(vr, vi, lane, wv_rot[5]);
  rz4<2>(vr, vi, lane, wv_rot[6]);  rx4<2>(vr, vi, wv_rot[7]);  rz4<2>(vr, vi, lane, wv_rot[8]);
  rz4<3>(vr, vi, lane, wv_rot[9]);  rx4<3>(vr, vi, wv_rot[10]); rz4<3>(vr, vi, lane, wv_rot[11]);
  crx4<0, 1>(vr, vi, lane, wv_crx[0]);
  crx4<1, 2>(vr, vi, lane, wv_crx[1]);
  crx4<2, 3>(vr, vi, lane, wv_crx[2]);
  crx4<3, 0>(vr, vi, lane, wv_crx[3]);
  cnot4<0, 1>(vr, vi, lane); cnot4<1, 2>(vr, vi, lane);
  cnot4<2, 3>(vr, vi, lane); cnot4<3, 0>(vr, vi, lane);
  cnot4<0, 3>(vr, vi, lane); cnot4<1, 0>(vr, vi, lane);
  cnot4<2, 1>(vr, vi, lane); cnot4<3, 2>(vr, vi, lane);
  rz4<0>(vr, vi, lane, wv_rot[12]); rx4<0>(vr, vi, wv_rot[13]); rz4<0>(vr, vi, lane, wv_rot[14]);
  rz4<1>(vr, vi, lane, wv_rot[15]); rx4<1>(vr, vi, wv_rot[16]); rz4<1>(vr, vi, lane, wv_rot[17]);
  rz4<2>(vr, vi, lane, wv_rot[18]); rx4<2>(vr, vi, wv_rot[19]); rz4<2>(vr, vi, lane, wv_rot[20]);
  rz4<3>(vr, vi, lane, wv_rot[21]); rx4<3>(vr, vi, wv_rot[22]); rz4<3>(vr, vi, lane, wv_rot[23]);
  // RX(tanh(mag)*pi)
  rx4<0>(vr, vi, tanhf(mag[0]) * PI_F);
  rx4<1>(vr, vi, tanhf(mag[1]) * PI_F);
  rx4<2>(vr, vi, tanhf(mag[2]) * PI_F);
  rx4<3>(vr, vi, tanhf(mag[3]) * PI_F);
  // CNOT ring
  cnot4<0, 1>(vr, vi, lane); cnot4<1, 2>(vr, vi, lane);
  cnot4<2, 3>(vr, vi, lane); cnot4<3, 0>(vr, vi, lane);

  float oz[4], ox[4];
  expzx4<0>(vr, vi, lane, oz[0], ox[0]);
  expzx4<1>(vr, vi, lane, oz[1], ox[1]);
  expzx4<2>(vr, vi, lane, oz[2], ox[2]);
  expzx4<3>(vr, vi, lane, oz[3], ox[3]);

  if (lane == 0) {
    float4* o = (float4*)(out + (size_t)wv * 8);
    o[0] = make_float4(oz[0], oz[1], oz[2], oz[3]);
    o[1] = make_float4(ox[0], ox[1], ox[2], ox[3]);
  }
}

// ---------------------------------------------------------------------------
extern "C" void kernel_launch(void* const* d_in, const int* in_sizes, int n_in,
                              void* d_out, int out_size, void* d_ws, size_t ws_size,
                              hipStream_t stream) {
  const float* x1     = (const float*)d_in[0];
  const float* x2     = (const float*)d_in[1];
  const float* W_pre  = (const float*)d_in[2];
  const float* b_pre  = (const float*)d_in[3];
  const float* wq_rot = (const float*)d_in[4];
  const float* wq_crx = (const float*)d_in[5];
  const float* wk_rot = (const float*)d_in[6];
  const float* wk_crx = (const float*)d_in[7];
  const float* wv_rot = (const float*)d_in[8];
  const float* wv_crx = (const float*)d_in[9];
  float* out = (float*)d_out;

  int B = in_sizes[0] / 96;
  float* x1p = (float*)d_ws;
  float* x2p = x1p + (size_t)B * 4;

  int tiles      = (B + 15) / 16;       // one wave per 16-row tile
  int pre_blocks = (tiles + 7) / 8;     // 8 waves / 256-thread block
  qattn_prenet_wmma<<<pre_blocks, 256, 0, stream>>>(x1, W_pre, b_pre, x1p, B);
  qattn_prenet_wmma<<<pre_blocks, 256, 0, stream>>>(x2, W_pre, b_pre, x2p, B);

  int sim_blocks = (B + 7) / 8;         // one wave per batch element
  qattn_sim<<<sim_blocks, 256, 0, stream>>>(x1p, x2p, wq_rot, wq_crx,
                                            wk_rot, wk_crx, wv_rot, wv_crx,
                                            out, B);
}